// SumPool_6305011990998
// MI455X (gfx1250) — compile-verified
//
#include <hip/hip_runtime.h>
#include <math.h>

// Problem constants (match reference)
#define B_MOL   128
#define A_ATOMS 64
#define N_ATOMS (B_MOL * A_ATOMS)   // 8192
#define K_NBR   32
#define P_PAIRS (N_ATOMS * K_NBR)   // 262144
#define D_HID   256

typedef float v2f __attribute__((ext_vector_type(2)));
typedef float v8f __attribute__((ext_vector_type(8)));

// Branchless tanh: 1 - 2/(exp2(2*log2e*x)+1).
// Lowers to v_exp_f32 + v_add + v_rcp_f32 + v_fma; saturates to +/-1 at the
// extremes (exp2->inf => +1, exp2->0 => -1); keeps EXEC all-ones (WMMA req).
__device__ __forceinline__ float fast_tanh(float x) {
    const float e = __builtin_amdgcn_exp2f(x * 2.8853900817779268f);  // 2/ln2
    const float r = __builtin_amdgcn_rcpf(e + 1.0f);
    return __builtin_fmaf(-2.0f, r, 1.0f);
}

// ---------------------------------------------------------------------------
// Zero the accumulators we atomically add into (d_out energy + d_ws stress).
// ---------------------------------------------------------------------------
__global__ void init_kernel(float* __restrict__ energy, float* __restrict__ ws_stress) {
    int i = blockIdx.x * blockDim.x + threadIdx.x;
    if (i < B_MOL)      energy[i]    = 0.0f;
    if (i < B_MOL * 9)  ws_stress[i] = 0.0f;
}

// ---------------------------------------------------------------------------
// Pair kernel: per tile of 16 pairs (one wave):
//   pre = R(16x3) @ W_pair(3x256) via 16x V_WMMA_F32_16X16X4_F32 (K padded to 4)
//   t = tanh(pre); phi = t.w2 summed per molecule;
//   outer = (W_pair @ (w2*(1-t^2))) (x) r summed per molecule (stress numerator)
// ---------------------------------------------------------------------------
__global__ __launch_bounds__(256)
void pair_kernel(const float* __restrict__ r_ij,
                 const long long* __restrict__ nbrs,   // (P,2) int64
                 const float* __restrict__ W_pair,     // (3,256)
                 const float* __restrict__ w2,         // (256,)
                 float* __restrict__ energy,           // (128,)
                 float* __restrict__ ws_stress) {      // (128,9)
    __shared__ float sW[3 * D_HID];
    __shared__ float sw2[D_HID];
    const int tid = threadIdx.x;
    for (int i = tid; i < 3 * D_HID; i += blockDim.x) sW[i]  = W_pair[i];
    for (int i = tid; i < D_HID;     i += blockDim.x) sw2[i] = w2[i];
    __syncthreads();

    const int  lane = tid & 31;
    const int  wave = tid >> 5;
    const bool hi   = lane >= 16;
    const int  lr   = lane & 15;

    const int tile = blockIdx.x * 8 + wave;   // 16384 tiles total
    const int p0   = tile * 16;
    const int q    = p0 + lr;                 // pair this lane stages for A

    const float rx = r_ij[3 * q + 0];
    const float ry = r_ij[3 * q + 1];
    const float rz = r_ij[3 * q + 2];

    // A matrix (16x4 f32): lanes0-15: VGPR0=K0(x), VGPR1=K1(y); lanes16-31: K2(z), K3(=0)
    v2f a;
    a.x = hi ? rz  : rx;
    a.y = hi ? 0.f : ry;

    // Broadcast the r-vectors of the 8 rows this lane half owns in D.
    float rvx[8], rvy[8], rvz[8];
    const int rbase = hi ? 8 : 0;
#pragma unroll
    for (int v = 0; v < 8; ++v) {
        rvx[v] = __shfl(rx, rbase + v, 32);
        rvy[v] = __shfl(ry, rbase + v, 32);
        rvz[v] = __shfl(rz, rbase + v, 32);
    }

    float phi_acc = 0.0f;
    float oacc[9] = {0, 0, 0, 0, 0, 0, 0, 0, 0};

    for (int nt = 0; nt < 16; ++nt) {
        const int n = nt * 16 + lr;           // hidden dim owned by this lane in D
        const float w0  = sW[0 * D_HID + n];
        const float w1v = sW[1 * D_HID + n];
        const float w2v = sW[2 * D_HID + n];
        const float wq  = sw2[n];

        // B matrix (4x16 f32): lanes0-15: VGPR0=K0, VGPR1=K1; lanes16-31: K2, K3(=0)
        v2f b;
        b.x = hi ? w2v : w0;
        b.y = hi ? 0.f : w1v;

        v8f c = {};
        c = __builtin_amdgcn_wmma_f32_16x16x4_f32(
                /*neg_a=*/false, a, /*neg_b=*/false, b,
                /*c_mod=*/(short)0, c, /*reuse_a=*/false, /*reuse_b=*/false);

        // D layout: VGPR v -> row M = v + (hi?8:0) (pair p0+M), N = lr (dim n)
#pragma unroll
        for (int v = 0; v < 8; ++v) {
            const float t = fast_tanh(c[v]);
            phi_acc += t * wq;
            const float s  = wq * (1.0f - t * t);
            const float g0 = w0  * s;
            const float g1 = w1v * s;
            const float g2 = w2v * s;
            oacc[0] += g0 * rvx[v]; oacc[1] += g0 * rvy[v]; oacc[2] += g0 * rvz[v];
            oacc[3] += g1 * rvx[v]; oacc[4] += g1 * rvy[v]; oacc[5] += g1 * rvz[v];
            oacc[6] += g2 * rvx[v]; oacc[7] += g2 * rvy[v]; oacc[8] += g2 * rvz[v];
        }
    }

    // Full 32-lane butterfly reduction of the 10 tile sums.
#pragma unroll
    for (int m = 16; m >= 1; m >>= 1) {
        phi_acc += __shfl_xor(phi_acc, m, 32);
#pragma unroll
        for (int i = 0; i < 9; ++i) oacc[i] += __shfl_xor(oacc[i], m, 32);
    }

    if (lane == 0) {
        const long long atom = nbrs[2 * (long long)p0];  // nbrs[:,0] = segment id
        const int mol = (int)(atom >> 6);                // 64 atoms per molecule
        atomicAdd(&energy[mol], phi_acc);
#pragma unroll
        for (int i = 0; i < 9; ++i) atomicAdd(&ws_stress[mol * 9 + i], oacc[i]);
    }
}

// ---------------------------------------------------------------------------
// Atom kernel: e_self contribution to energy + g_xyz output (tiny workload).
// ---------------------------------------------------------------------------
__global__ __launch_bounds__(256)
void atom_kernel(const float* __restrict__ xyz,
                 const float* __restrict__ W_self,   // (3,256)
                 const float* __restrict__ w1,       // (256,)
                 float* __restrict__ energy,
                 float* __restrict__ g_xyz) {
    __shared__ float sW[3 * D_HID];
    __shared__ float sw1[D_HID];
    const int tid = threadIdx.x;
    for (int i = tid; i < 3 * D_HID; i += blockDim.x) sW[i]  = W_self[i];
    for (int i = tid; i < D_HID;     i += blockDim.x) sw1[i] = w1[i];
    __syncthreads();

    const int n = blockIdx.x * blockDim.x + tid;   // 8192 atoms
    const float x = xyz[3 * n + 0];
    const float y = xyz[3 * n + 1];
    const float z = xyz[3 * n + 2];

    float e = 0.f, gx = 0.f, gy = 0.f, gz = 0.f;
    for (int d = 0; d < D_HID; ++d) {
        const float w0  = sW[0 * D_HID + d];
        const float w1r = sW[1 * D_HID + d];
        const float w2r = sW[2 * D_HID + d];
        const float pre = x * w0 + y * w1r + z * w2r;
        const float t   = fast_tanh(pre);
        const float wq  = sw1[d];
        e += t * wq;
        const float s = wq * (1.0f - t * t);
        gx += w0 * s; gy += w1r * s; gz += w2r * s;
    }
    g_xyz[3 * n + 0] = gx;
    g_xyz[3 * n + 1] = gy;
    g_xyz[3 * n + 2] = gz;

    // Wave spans 32 consecutive atoms, always inside one 64-atom molecule.
#pragma unroll
    for (int m = 16; m >= 1; m >>= 1) e += __shfl_xor(e, m, 32);
    if ((tid & 31) == 0) atomicAdd(&energy[n >> 6], e);
}

// ---------------------------------------------------------------------------
// Finalize stress: divide per-molecule outer-product sums by |det(cell_b)|.
// ---------------------------------------------------------------------------
__global__ void stress_kernel(const float* __restrict__ cell,       // (3B,3)
                              const float* __restrict__ ws_stress,  // (B,9)
                              float* __restrict__ stress_out) {     // (3B,3)
    const int b = blockIdx.x * blockDim.x + threadIdx.x;
    if (b >= B_MOL) return;
    const float* c = cell + 9 * b;
    const float det = c[0] * (c[4] * c[8] - c[5] * c[7])
                    - c[1] * (c[3] * c[8] - c[5] * c[6])
                    + c[2] * (c[3] * c[7] - c[4] * c[6]);
    const float inv = 1.0f / fabsf(det);
#pragma unroll
    for (int i = 0; i < 9; ++i) stress_out[9 * b + i] = ws_stress[9 * b + i] * inv;
}

// ---------------------------------------------------------------------------
extern "C" void kernel_launch(void* const* d_in, const int* in_sizes, int n_in,
                              void* d_out, int out_size, void* d_ws, size_t ws_size,
                              hipStream_t stream) {
    (void)in_sizes; (void)n_in; (void)out_size; (void)ws_size;

    const float*     xyz    = (const float*)d_in[0];
    const float*     r_ij   = (const float*)d_in[1];
    const long long* nbrs   = (const long long*)d_in[2];
    const float*     cell   = (const float*)d_in[3];
    const float*     W_self = (const float*)d_in[4];
    const float*     W_pair = (const float*)d_in[5];
    const float*     w1     = (const float*)d_in[6];
    const float*     w2     = (const float*)d_in[7];

    float* out    = (float*)d_out;
    float* energy = out;                        // 128
    float* g_xyz  = out + B_MOL;                // 8192*3
    float* stress = out + B_MOL + 3 * N_ATOMS;  // 384*3
    float* ws_st  = (float*)d_ws;               // 128*9 floats scratch

    init_kernel<<<5, 256, 0, stream>>>(energy, ws_st);

    // 16384 pair-tiles, 8 waves (tiles) per 256-thread block -> 2048 blocks
    pair_kernel<<<P_PAIRS / 16 / 8, 256, 0, stream>>>(r_ij, nbrs, W_pair, w2,
                                                      energy, ws_st);

    atom_kernel<<<N_ATOMS / 256, 256, 0, stream>>>(xyz, W_self, w1, energy, g_xyz);

    stress_kernel<<<1, 128, 0, stream>>>(cell, ws_st, stress);
}